// IsoMaxPlusLossFirstPart_37495064494857
// MI455X (gfx1250) — compile-verified
//
#include <hip/hip_runtime.h>

typedef __attribute__((ext_vector_type(2))) float v2f;
typedef __attribute__((ext_vector_type(8))) float v8f;
typedef __attribute__((ext_vector_type(4))) int   v4i;

#define DVEC 512
#define BM 128
#define BN 64
#define BK 32
#define NCHUNK (DVEC / BK)     // 16
#define LDSS 36                // BK + 4 pad: conflict-free 16x4 f32 fragment reads
#define ASYNC_PER_CHUNK 6      // per-wave: 4 A-tile + 2 B-tile async b128 issues

typedef __attribute__((address_space(1))) v4i* gv4_t;   // global int4*
typedef __attribute__((address_space(3))) v4i* lv4_t;   // LDS int4*

// ---- CDNA5 async global->LDS (ASYNCcnt) path ----
#if defined(__has_builtin) && __has_builtin(__builtin_amdgcn_global_load_async_to_lds_b128)
#define ASYNC_COPY_B128(gp, lp) \
    __builtin_amdgcn_global_load_async_to_lds_b128((gv4_t)(gp), (lv4_t)(lp), 0, 0)
#else
#define ASYNC_COPY_B128(gp, lp)                                                   \
    do {                                                                          \
        unsigned _lo = (unsigned)(unsigned long long)(lv4_t)(lp);                 \
        unsigned long long _ga = (unsigned long long)(gp);                        \
        asm volatile("global_load_async_to_lds_b128 %0, %1, off"                  \
                     :: "v"(_lo), "v"(_ga) : "memory");                           \
    } while (0)
#endif

#if defined(__has_builtin) && __has_builtin(__builtin_amdgcn_s_wait_asynccnt)
#define WAIT_ASYNC(n) __builtin_amdgcn_s_wait_asynccnt(n)
#else
#define WAIT_ASYNC(n) asm volatile("s_wait_asynccnt %0" :: "i"(n) : "memory")
#endif

__global__ __launch_bounds__(256) void rownorm_kernel(const float* __restrict__ x,
                                                      float* __restrict__ inv,
                                                      int rows) {
    const int lane = threadIdx.x & 31;
    const int wave = threadIdx.x >> 5;
    const int row  = blockIdx.x * 8 + wave;
    if (row >= rows) return;
    const float4* p = (const float4*)(x + (size_t)row * DVEC);
    float s = 0.f;
#pragma unroll
    for (int i = 0; i < 4; ++i) {
        float4 v = p[lane + i * 32];
        s += v.x * v.x + v.y * v.y + v.z * v.z + v.w * v.w;
    }
#pragma unroll
    for (int off = 16; off > 0; off >>= 1) s += __shfl_xor(s, off, 32);
    if (lane == 0) inv[row] = 1.0f / fmaxf(sqrtf(s), 1e-12f);
}

__global__ __launch_bounds__(256) void isomax_gemm_kernel(
    const float* __restrict__ feat, const float* __restrict__ proto,
    const float* __restrict__ scale_p,
    const float* __restrict__ finv, const float* __restrict__ pinv,
    float* __restrict__ out, int Bn, int C) {
    __shared__ float Asl[2][BM * LDSS];
    __shared__ float Bsl[2][BN * LDSS];

    const int tid  = threadIdx.x;
    const int lane = tid & 31;
    const int wave = tid >> 5;
    const int m0 = blockIdx.x * BM;
    const int n0 = blockIdx.y * BN;

    // 8 waves = 4 (M) x 2 (N); each wave owns a 32x32 output tile (2x2 WMMA tiles)
    const int wm = wave >> 1;
    const int wn = wave & 1;

    v8f acc00 = {}, acc01 = {}, acc10 = {}, acc11 = {};

    // 16x4 f32 fragment addressing: lanes 0-15 hold K=k,k+1; lanes 16-31 hold K=k+2,k+3
    const int lrow  = lane & 15;
    const int khalf = (lane >> 4) * 2;
    const int mrow0 = wm * 32 + lrow;
    const int nrow0 = wn * 32 + lrow;

    // cooperative async copy: 8 float4 per row, thread t -> row (t>>3), col4 (t&7)*4
    const int lc4 = (tid & 7) * 4;
    const int lr  = tid >> 3;   // 0..31

    // prologue: chunk 0 -> buffer 0
    {
#pragma unroll
        for (int r = 0; r < BM; r += 32) {
            const int grow = m0 + lr + r;
            if (grow < Bn)
                ASYNC_COPY_B128(feat + (size_t)grow * DVEC + lc4,
                                &Asl[0][(lr + r) * LDSS + lc4]);
        }
#pragma unroll
        for (int r = 0; r < BN; r += 32) {
            const int prow = n0 + lr + r;
            if (prow < C)
                ASYNC_COPY_B128(proto + (size_t)prow * DVEC + lc4,
                                &Bsl[0][(lr + r) * LDSS + lc4]);
        }
    }

    for (int c = 0; c < NCHUNK; ++c) {
        if (c + 1 < NCHUNK) {
            const int kc = (c + 1) * BK;
            const int nb = (c + 1) & 1;
#pragma unroll
            for (int r = 0; r < BM; r += 32) {
                const int grow = m0 + lr + r;
                if (grow < Bn)
                    ASYNC_COPY_B128(feat + (size_t)grow * DVEC + kc + lc4,
                                    &Asl[nb][(lr + r) * LDSS + lc4]);
            }
#pragma unroll
            for (int r = 0; r < BN; r += 32) {
                const int prow = n0 + lr + r;
                if (prow < C)
                    ASYNC_COPY_B128(proto + (size_t)prow * DVEC + kc + lc4,
                                    &Bsl[nb][(lr + r) * LDSS + lc4]);
            }
            WAIT_ASYNC(ASYNC_PER_CHUNK);  // chunk c complete (in-order retirement)
        } else {
            WAIT_ASYNC(0);
        }
        __syncthreads();

        const float* As = Asl[c & 1];
        const float* Bs = Bsl[c & 1];
#pragma unroll
        for (int k = 0; k < BK; k += 4) {
            v2f a0 = *(const v2f*)&As[mrow0 * LDSS + k + khalf];
            v2f a1 = *(const v2f*)&As[(mrow0 + 16) * LDSS + k + khalf];
            v2f b0 = *(const v2f*)&Bs[nrow0 * LDSS + k + khalf];
            v2f b1 = *(const v2f*)&Bs[(nrow0 + 16) * LDSS + k + khalf];
            acc00 = __builtin_amdgcn_wmma_f32_16x16x4_f32(false, a0, false, b0,
                                                          (short)0, acc00, false, false);
            acc01 = __builtin_amdgcn_wmma_f32_16x16x4_f32(false, a0, false, b1,
                                                          (short)0, acc01, false, false);
            acc10 = __builtin_amdgcn_wmma_f32_16x16x4_f32(false, a1, false, b0,
                                                          (short)0, acc10, false, false);
            acc11 = __builtin_amdgcn_wmma_f32_16x16x4_f32(false, a1, false, b1,
                                                          (short)0, acc11, false, false);
        }
        __syncthreads();
    }

    // epilogue: C/D layout -> VGPR v, lanes 0-15: M=v, N=lane; lanes 16-31: M=v+8, N=lane-16
    const float sc   = fabsf(scale_p[0]);
    const int colA   = n0 + wn * 32 + lrow;
    const int colB   = colA + 16;
    const int rbase  = m0 + wm * 32 + ((lane >> 4) * 8);
    const float pvA  = (colA < C) ? pinv[colA] : 0.f;
    const float pvB  = (colB < C) ? pinv[colB] : 0.f;

#pragma unroll
    for (int v = 0; v < 8; ++v) {
        const int ra = rbase + v;
        if (ra < Bn) {
            const float fa = finv[ra];
            if (colA < C)
                out[(size_t)ra * C + colA] =
                    -sc * sqrtf(fmaxf(2.f - 2.f * (acc00[v] * fa * pvA), 1e-12f));
            if (colB < C)
                out[(size_t)ra * C + colB] =
                    -sc * sqrtf(fmaxf(2.f - 2.f * (acc01[v] * fa * pvB), 1e-12f));
        }
        const int rb = rbase + 16 + v;
        if (rb < Bn) {
            const float fb = finv[rb];
            if (colA < C)
                out[(size_t)rb * C + colA] =
                    -sc * sqrtf(fmaxf(2.f - 2.f * (acc10[v] * fb * pvA), 1e-12f));
            if (colB < C)
                out[(size_t)rb * C + colB] =
                    -sc * sqrtf(fmaxf(2.f - 2.f * (acc11[v] * fb * pvB), 1e-12f));
        }
    }
}

extern "C" void kernel_launch(void* const* d_in, const int* in_sizes, int n_in,
                              void* d_out, int out_size, void* d_ws, size_t ws_size,
                              hipStream_t stream) {
    const float* feat  = (const float*)d_in[0];  // [B, 512]
    const float* proto = (const float*)d_in[1];  // [C, 512]
    const float* scale = (const float*)d_in[2];  // [1]
    const int Bn = in_sizes[0] / DVEC;           // 65536
    const int C  = in_sizes[1] / DVEC;           // 1000

    float* finv = (float*)d_ws;                  // Bn floats
    float* pinv = finv + Bn;                     // C floats

    rownorm_kernel<<<(Bn + 7) / 8, 256, 0, stream>>>(feat, finv, Bn);
    rownorm_kernel<<<(C + 7) / 8, 256, 0, stream>>>(proto, pinv, C);

    dim3 grid((Bn + BM - 1) / BM, (C + BN - 1) / BN);
    isomax_gemm_kernel<<<grid, 256, 0, stream>>>(feat, proto, scale, finv, pinv,
                                                 (float*)d_out, Bn, C);
}